// RetinaNetClassificationLoss_12893491822713
// MI455X (gfx1250) — compile-verified
//
#include <hip/hip_runtime.h>

typedef float v2f __attribute__((ext_vector_type(2)));
typedef float v8f __attribute__((ext_vector_type(8)));

#define WPI 250             // workgroups per image
#define WAVES_PER_WG 8
#define TPB (WAVES_PER_WG * 32)

// ---------------------------------------------------------------------------
// Kernel 1: per-workgroup partial focal-loss sum + foreground count.
// One wave owns a contiguous run of anchors; lanes stride the class dim
// (coalesced 128B bursts). Inner loop is fully branchless (v_cndmask + FMA).
// Wave reduction uses V_WMMA_F32_16X16X4_F32 with an all-ones B matrix:
//   D[m][n] = sum_k A[m][k]  (C = 0)
// => sum over ALL lane-registers of D == 16 * sum over ALL elements of A,
// valid for ANY bijective lane<->element striping. A = {partial, 0} per lane.
// ---------------------------------------------------------------------------
__global__ __launch_bounds__(TPB) void focal_partial_kernel(
    const float* __restrict__ logits,   // [B, A, C]
    const int*   __restrict__ labels,   // [B, M]
    const int*   __restrict__ matched,  // [B, A]  in {-2,-1,0..M-1}
    float*       __restrict__ part_loss,
    int*         __restrict__ part_cnt,
    int A, int C, int M)
{
    const int b    = blockIdx.x / WPI;
    const int wgi  = blockIdx.x % WPI;
    const int wave = threadIdx.x >> 5;
    const int lane = threadIdx.x & 31;

    const int waves_per_img = WPI * WAVES_PER_WG;
    const int wave_img      = wgi * WAVES_PER_WG + wave;
    const int per_wave      = (A + waves_per_img - 1) / waves_per_img;
    const int a0            = wave_img * per_wave;
    const int a1            = min(A, a0 + per_wave);

    const float* __restrict__ lg = logits  + (size_t)b * A * C;
    const int*   __restrict__ mi = matched + (size_t)b * A;
    const int*   __restrict__ lb = labels  + (size_t)b * M;

    float acc = 0.0f;   // per-lane loss partial
    int   cnt = 0;      // foreground anchors (wave-uniform)

    for (int a = a0; a < a1; ++a) {
        const int m = mi[a];               // broadcast load, wave-uniform
        if (m == -2) continue;             // between thresholds: ignored
        const bool fg  = (m >= 0);
        cnt += fg ? 1 : 0;
        const int  cls = fg ? lb[m] : -1;  // gathered GT class (else no match)
        const float* __restrict__ row = lg + (size_t)a * C;

        for (int c = lane; c < C; c += 32) {
            const float x    = row[c];
            const float e    = __expf(-fabsf(x));    // exp(-|x|) <= 1
            const float inv  = 1.0f + e;
            const float r    = 1.0f / inv;           // 1/(1+e)
            const float q    = e * r;                // e/(1+e)
            const float ce0  = fmaxf(x, 0.0f) + __logf(inv); // BCE, t=0
            const bool  xpos = (x >= 0.0f);
            const bool  is_t = (c == cls);           // fg baked into cls==-1
            // branchless focal loss (GAMMA=2 -> square):
            //   sel     = is_t ? 1 : 0
            //   alpha_t = 0.75 - 0.5*sel
            //   ce      = ce0 - sel*x
            //   f       = is_t ? (1-p) : p  ==  (xpos==is_t) ? q : r
            const float sel   = is_t ? 1.0f : 0.0f;
            const float alpha = __builtin_fmaf(-0.5f, sel, 0.75f);
            const float ce    = __builtin_fmaf(-sel, x, ce0);
            const float f     = (xpos == is_t) ? q : r;
            acc = __builtin_fmaf(alpha * ce, f * f, acc);
        }
    }

    // --- wave reduction via WMMA (EXEC all-1s: converged uniform flow) ---
    v2f Amat; Amat[0] = acc;  Amat[1] = 0.0f;
    v2f Bmat; Bmat[0] = 1.0f; Bmat[1] = 1.0f;
    v8f Cz = {};
    v8f D = __builtin_amdgcn_wmma_f32_16x16x4_f32(
        /*neg_a=*/false, Amat, /*neg_b=*/false, Bmat,
        /*c_mod=*/(short)0, Cz, /*reuse_a=*/false, /*reuse_b=*/false);

    float s = D[0] + D[1] + D[2] + D[3] + D[4] + D[5] + D[6] + D[7];
#pragma unroll
    for (int off = 16; off >= 1; off >>= 1)
        s += __shfl_xor(s, off, 32);
    const float wave_sum = s * (1.0f / 16.0f);   // undo the 16 column copies

    __shared__ float wsum[WAVES_PER_WG];
    __shared__ int   wcnt[WAVES_PER_WG];
    if (lane == 0) { wsum[wave] = wave_sum; wcnt[wave] = cnt; }
    __syncthreads();

    if (threadIdx.x == 0) {                      // fixed order: deterministic
        float t = 0.0f; int n = 0;
#pragma unroll
        for (int i = 0; i < WAVES_PER_WG; ++i) { t += wsum[i]; n += wcnt[i]; }
        part_loss[blockIdx.x] = t;
        part_cnt[blockIdx.x]  = n;
    }
}

// ---------------------------------------------------------------------------
// Kernel 2: one wave. Lane b (<B) serially folds its image's WPI partials in
// fixed order, normalizes by max(1, num_fg), then an 8-wide shfl tree
// averages the per-image losses. Fully deterministic.
// ---------------------------------------------------------------------------
__global__ __launch_bounds__(32) void focal_final_kernel(
    const float* __restrict__ part_loss,
    const int*   __restrict__ part_cnt,
    float*       __restrict__ out, int B)
{
    const int lane = threadIdx.x;
    float per = 0.0f;
    if (lane < B) {
        float s = 0.0f; int n = 0;
        for (int i = 0; i < WPI; ++i) {
            s += part_loss[lane * WPI + i];
            n += part_cnt[lane * WPI + i];
        }
        per = s / (float)max(1, n);
    }
#pragma unroll
    for (int off = 4; off >= 1; off >>= 1)      // lanes 8..31 hold 0
        per += __shfl_xor(per, off, 32);
    if (lane == 0) out[0] = per / (float)B;
}

// ---------------------------------------------------------------------------
extern "C" void kernel_launch(void* const* d_in, const int* in_sizes, int n_in,
                              void* d_out, int out_size, void* d_ws, size_t ws_size,
                              hipStream_t stream)
{
    const float* logits  = (const float*)d_in[0];  // [B, A, C] f32
    const int*   labels  = (const int*)  d_in[1];  // [B, M]
    const int*   matched = (const int*)  d_in[2];  // [B, A]

    const int B = 8;
    const int M = in_sizes[1] / B;                 // 100
    const int A = in_sizes[2] / B;                 // 80000
    const int C = in_sizes[0] / in_sizes[2];       // 91

    const int G = B * WPI;                         // 2000 partials
    float* part_loss = (float*)d_ws;
    int*   part_cnt  = (int*)((char*)d_ws + sizeof(float) * (size_t)G);

    focal_partial_kernel<<<G, TPB, 0, stream>>>(logits, labels, matched,
                                                part_loss, part_cnt, A, C, M);
    focal_final_kernel<<<1, 32, 0, stream>>>(part_loss, part_cnt,
                                             (float*)d_out, B);
}